// TransposableSparse_71932112273438
// MI455X (gfx1250) — compile-verified
//
#include <hip/hip_runtime.h>

typedef float v2f __attribute__((ext_vector_type(2)));
typedef float v8f __attribute__((ext_vector_type(8)));
typedef unsigned long long u64;
typedef unsigned int u32;

// 90 transposable 2:4 patterns over a flattened 4x4 tile. String char e == bit (15-e)
// of the binary literal, so pattern[p][e] = (PBITS[p] >> (15 - e)) & 1.
// Padded to 96 with zeros (zero score never beats a real pattern; all-zero ties
// resolve to index 0, matching jnp.argmax first-max semantics).
__constant__ unsigned short PBITS[96] = {
  0b1100110000110011, 0b1100101001010011, 0b1100101000110101, 0b1100100101100011, 0b1100100100110110,
  0b1100011010010011, 0b1100011000111001, 0b1100010110100011, 0b1100010100111010, 0b1100001111000011,
  0b1100001110100101, 0b1100001110010110, 0b1100001101101001, 0b1100001101011010, 0b1100001100111100,
  0b1010110001010011, 0b1010110000110101, 0b1010101001010101, 0b1010100101100101, 0b1010100101010110,
  0b1010011010010101, 0b1010011001011001, 0b1010010111000011, 0b1010010110100101, 0b1010010110010110,
  0b1010010101101001, 0b1010010101011010, 0b1010010100111100, 0b1010001111000101, 0b1010001101011100,
  0b1001110001100011, 0b1001110000110110, 0b1001101001100101, 0b1001101001010110, 0b1001100101100110,
  0b1001011011000011, 0b1001011010100101, 0b1001011010010110, 0b1001011001101001, 0b1001011001011010,
  0b1001011000111100, 0b1001010110100110, 0b1001010101101010, 0b1001001111000110, 0b1001001101101100,
  0b0110110010010011, 0b0110110000111001, 0b0110101010010101, 0b0110101001011001, 0b0110100111000011,
  0b0110100110100101, 0b0110100110010110, 0b0110100101101001, 0b0110100101011010, 0b0110100100111100,
  0b0110011010011001, 0b0110010110101001, 0b0110010110011010, 0b0110001111001001, 0b0110001110011100,
  0b0101110010100011, 0b0101110000111010, 0b0101101011000011, 0b0101101010100101, 0b0101101010010110,
  0b0101101001101001, 0b0101101001011010, 0b0101101000111100, 0b0101100110100110, 0b0101100101101010,
  0b0101011010101001, 0b0101011010011010, 0b0101010110101010, 0b0101001111001010, 0b0101001110101100,
  0b0011110011000011, 0b0011110010100101, 0b0011110010010110, 0b0011110001101001, 0b0011110001011010,
  0b0011110000111100, 0b0011101011000101, 0b0011101001011100, 0b0011100111000110, 0b0011100101101100,
  0b0011011011001001, 0b0011011010011100, 0b0011010111001010, 0b0011010110101100, 0b0011001111001100,
  0, 0, 0, 0, 0, 0
};

#define GPW 8          // tile-groups (of 16 tiles) per wave
#define KDIM 8192      // reference matrix: 4096 x 8192 f32
#define MDIM 4096

// One wave handles GPW groups; each group = 16 tiles (4 rows x 64 cols).
// Score GEMM: M=16 tiles, N=96 patterns (6 blocks of 16), K=16 tile elements via
// 4 chained V_WMMA_F32_16X16X4_F32 (exact f32 -> argmax matches the f32 reference).
// Argmax uses u64 keys: (f32 score bits << 32) | (127 - idx). Scores are >= 0, so
// float-bit order == uint order; u64 max == (max score, min index on ties).
__global__ __launch_bounds__(256) void tsparse24_kernel(const float* __restrict__ x,
                                                        float* __restrict__ outS,
                                                        float* __restrict__ outM) {
  const int lane   = (int)(threadIdx.x & 31);
  const int wave   = (int)((blockIdx.x * blockDim.x + threadIdx.x) >> 5);
  const int n      = lane & 15;       // tile index in group / pattern index in block
  const int half   = lane >> 4;       // K-pair split of the 16x4 f32 A layout
  const int colOff = half * 2;        // lanes 0-15: tile cols 0,1 ; lanes 16-31: cols 2,3

  const int wavesPerRow = (KDIM / 64) / GPW;       // 16
  const int tileRow = wave / wavesPerRow;
  const int seg     = wave % wavesPerRow;

  // B fragments (pattern matrix) built once per wave.
  // B layout for 4x16 f32: lanes 0-15 hold K=0,1 (vgpr0,1); lanes 16-31 hold K=2,3.
  v2f bfrag[6][4];
  u32 lowW[6];                         // key low word: 127 - pattern index
#pragma unroll
  for (int nb = 0; nb < 6; ++nb) {
    const unsigned bits = PBITS[nb * 16 + n];
    lowW[nb] = 127u - (u32)(nb * 16 + n);
#pragma unroll
    for (int j = 0; j < 4; ++j) {
      const int sh0 = 15 - (4 * j + colOff);
      bfrag[nb][j].x = (float)((bits >> sh0) & 1u);
      bfrag[nb][j].y = (float)((bits >> (sh0 - 1)) & 1u);
    }
  }

  const size_t rowBase = (size_t)(tileRow * 4) * (size_t)KDIM;

  for (int g = 0; g < GPW; ++g) {
    const int col = (seg * GPW + g) * 64 + n * 4 + colOff;
    const float* px = x + rowBase + col;
    if (g + 1 < GPW) __builtin_prefetch(px + 64, 0, 3);   // next group, near scope

    // A data: 4 tile rows, float2 per lane per row (wave covers 256B/row, coalesced).
    v2f xv[4], a[4];
#pragma unroll
    for (int j = 0; j < 4; ++j) {
      xv[j] = *(const v2f*)(px + (size_t)j * KDIM);
      a[j].x = __builtin_fabsf(xv[j].x);
      a[j].y = __builtin_fabsf(xv[j].y);
    }

    u64 bestKey[8];
#pragma unroll
    for (int v = 0; v < 8; ++v) bestKey[v] = 0ull;   // any real key beats 0 (low word >= 32)

#pragma unroll
    for (int nb = 0; nb < 6; ++nb) {
      v8f c = {0.f, 0.f, 0.f, 0.f, 0.f, 0.f, 0.f, 0.f};
#pragma unroll
      for (int j = 0; j < 4; ++j)
        c = __builtin_amdgcn_wmma_f32_16x16x4_f32(false, a[j], false, bfrag[nb][j],
                                                  (short)0, c, false, false);
#pragma unroll
      for (int v = 0; v < 8; ++v) {
        const u64 key = ((u64)__float_as_uint(c[v]) << 32) | (u64)lowW[nb];
        if (key > bestKey[v]) bestKey[v] = key;      // ascending idx -> '>' keeps first max
      }
    }

    // Cross-lane argmax per C row: u64-max butterfly within each 16-lane half
    // (masks 1..8 never cross bit 4). Lanes 0-15 end with tile v, lanes 16-31 tile 8+v.
    u32 lo[8];
#pragma unroll
    for (int v = 0; v < 8; ++v) {
      u64 key = bestKey[v];
#pragma unroll
      for (int mlane = 8; mlane >= 1; mlane >>= 1) {
        const u64 ok = __shfl_xor(key, mlane);
        if (ok > key) key = ok;
      }
      lo[v] = (u32)key;                              // only the index word is needed now
    }

    // Route tile n's winner to the lane owning tile n's data:
    // 1) per-lane select of register (lane & 7); 2) one bpermute to swap halves.
    u32 w = lo[0];
#pragma unroll
    for (int v = 1; v < 8; ++v) w = ((lane & 7) == v) ? lo[v] : w;
    const int srcLane = (lane & 7) | (((lane >> 3) & 1) << 4);
    const u32 wq = (u32)__shfl((int)w, srcLane);
    const int myBest = 127 - (int)wq;

    // Apply mask; stores are streaming (write-once) -> non-temporal.
    const unsigned bits = PBITS[myBest];
    float* ps = outS + rowBase + col;
    float* pm = outM + rowBase + col;
#pragma unroll
    for (int j = 0; j < 4; ++j) {
      const int sh0 = 15 - (4 * j + colOff);
      const float f0 = (float)((bits >> sh0) & 1u);
      const float f1 = (float)((bits >> (sh0 - 1)) & 1u);
      v2f s, mk;
      s.x = xv[j].x * f0; s.y = xv[j].y * f1;
      mk.x = f0;          mk.y = f1;
      __builtin_nontemporal_store(s,  (v2f*)(ps + (size_t)j * KDIM));
      __builtin_nontemporal_store(mk, (v2f*)(pm + (size_t)j * KDIM));
    }
  }
}

extern "C" void kernel_launch(void* const* d_in, const int* in_sizes, int n_in,
                              void* d_out, int out_size, void* d_ws, size_t ws_size,
                              hipStream_t stream) {
  (void)in_sizes; (void)n_in; (void)out_size; (void)d_ws; (void)ws_size;
  const float* x = (const float*)d_in[0];
  float* outS = (float*)d_out;
  float* outM = outS + (size_t)MDIM * (size_t)KDIM;   // outputs concatenated: sparse, mask

  const int totalWaves = (MDIM / 4) * ((KDIM / 64) / GPW);  // 1024 * 16 = 16384
  const int wavesPerBlock = 256 / 32;                       // 8
  dim3 grid(totalWaves / wavesPerBlock), block(256);
  tsparse24_kernel<<<grid, block, 0, stream>>>(x, outS, outM);
}